// OptimAttn_6459630813624
// MI455X (gfx1250) — compile-verified
//
#include <hip/hip_runtime.h>
#include <math.h>

#define BSZ  32
#define SZ   2048
#define OUTN 2048
#define LN_EPS 1e-5f

typedef float v2f __attribute__((ext_vector_type(2)));
typedef float v8f __attribute__((ext_vector_type(8)));

// ---------------------------------------------------------------------------
// Kernel 1: q/k/v = x @ W^T  using V_WMMA_F32_16X16X4_F32 (exact fp32 path).
// One wave per (weight, 16-wide N strip). Wave computes both M-tiles (M=32).
// A layout (16x4 f32, ISA 7.12.2): lane L: M=L&15, K = 2*(L>>4)+j  (j=VGPR idx)
// B layout (4x16 f32, mirrored):   lane L: N=L&15, K = 2*(L>>4)+j
// C/D layout: VGPR r -> row (r + (L>>4)*8), col = L&15
// ---------------------------------------------------------------------------
__global__ __launch_bounds__(128) void qkv_gemm(
    const float* __restrict__ x,
    const float* __restrict__ Wq,
    const float* __restrict__ Wk,
    const float* __restrict__ Wv,
    float* __restrict__ q, float* __restrict__ k, float* __restrict__ v)
{
    const int wave  = (blockIdx.x * blockDim.x + threadIdx.x) >> 5; // 0..383
    const int lane  = threadIdx.x & 31;
    const int which = wave >> 7;     // 0=q,1=k,2=v
    const int ntile = wave & 127;    // 0..127

    const float* __restrict__ W   = (which == 0) ? Wq : (which == 1) ? Wk : Wv;
    float*       __restrict__ out = (which == 0) ? q  : (which == 1) ? k  : v;

    const int m  = lane & 15;
    const int kh = lane >> 4;        // 0 or 1
    const int n  = lane & 15;

    const float* xr0 = x + m * SZ;          // M-tile 0 row
    const float* xr1 = x + (16 + m) * SZ;   // M-tile 1 row
    const float* wr  = W + (size_t)(ntile * 16 + n) * SZ;

    v8f c0 = {}; v8f c1 = {};
    #pragma unroll 8
    for (int k0 = 0; k0 < SZ; k0 += 4) {
        const int kk = k0 + kh * 2;
        v2f a0 = *(const v2f*)(xr0 + kk);
        v2f a1 = *(const v2f*)(xr1 + kk);
        v2f bb = *(const v2f*)(wr  + kk);
        c0 = __builtin_amdgcn_wmma_f32_16x16x4_f32(false, a0, false, bb,
                                                   (short)0, c0, false, false);
        c1 = __builtin_amdgcn_wmma_f32_16x16x4_f32(false, a1, false, bb,
                                                   (short)0, c1, false, false);
    }

    const int col = ntile * 16 + n;
    #pragma unroll
    for (int r = 0; r < 8; ++r) {
        out[(size_t)(r + kh * 8) * SZ + col]      = c0[r];
        out[(size_t)(16 + r + kh * 8) * SZ + col] = c1[r];
    }
}

// ---------------------------------------------------------------------------
// Kernel 2: flattened-softmax attention (never materializes [B,S,S]) fused
// with the OptimActivation. One workgroup per batch row.
// ---------------------------------------------------------------------------
__global__ __launch_bounds__(256) void attn_act(
    const float* __restrict__ q,
    const float* __restrict__ k,
    const float* __restrict__ v,
    const float* __restrict__ p2,
    float* __restrict__ fc)
{
    __shared__ float qs[SZ], ks[SZ], vs[SZ];
    __shared__ float accs[SZ];
    __shared__ float red[256];
    __shared__ float stat[5];   // qmax,qmin,kmax,kmin,Z

    const int b = blockIdx.x;
    const int t = threadIdx.x;

    for (int i = t; i < SZ; i += 256) {
        qs[i] = q[(size_t)b * SZ + i];
        ks[i] = k[(size_t)b * SZ + i];
        vs[i] = v[(size_t)b * SZ + i];
    }
    __syncthreads();

    float qmx = -3.0e38f, qmn = 3.0e38f, kmx = -3.0e38f, kmn = 3.0e38f;
    for (int i = t; i < SZ; i += 256) {
        qmx = fmaxf(qmx, qs[i]); qmn = fminf(qmn, qs[i]);
        kmx = fmaxf(kmx, ks[i]); kmn = fminf(kmn, ks[i]);
    }
    red[t] = qmx; __syncthreads();
    if (t == 0) { float r = red[0]; for (int i = 1; i < 256; ++i) r = fmaxf(r, red[i]); stat[0] = r; }
    __syncthreads();
    red[t] = qmn; __syncthreads();
    if (t == 0) { float r = red[0]; for (int i = 1; i < 256; ++i) r = fminf(r, red[i]); stat[1] = r; }
    __syncthreads();
    red[t] = kmx; __syncthreads();
    if (t == 0) { float r = red[0]; for (int i = 1; i < 256; ++i) r = fmaxf(r, red[i]); stat[2] = r; }
    __syncthreads();
    red[t] = kmn; __syncthreads();
    if (t == 0) { float r = red[0]; for (int i = 1; i < 256; ++i) r = fminf(r, red[i]); stat[3] = r; }
    __syncthreads();

    const float Qx = stat[0], Qn = stat[1], Kx = stat[2], Kn = stat[3];
    const float mb = fmaxf(fmaxf(Qx * Kx, Qx * Kn), fmaxf(Qn * Kx, Qn * Kn));

    float zpart = 0.0f;
    for (int i = t; i < SZ; i += 256) {
        const float qi = qs[i];
        float acc = 0.0f, rz = 0.0f;
        for (int j = 0; j < SZ; ++j) {
            const float e = expf(qi * ks[j] - mb);
            acc += e * vs[j];
            rz  += e;
        }
        accs[i] = acc;
        zpart  += rz;
    }
    red[t] = zpart; __syncthreads();
    if (t == 0) { float s = 0.0f; for (int i = 0; i < 256; ++i) s += red[i]; stat[4] = s; }
    __syncthreads();
    const float Zinv = 1.0f / stat[4];

    // softmax(p2) branch weights (5 params, computed per-thread — cheap)
    float pbuf[5];
    float pm = p2[0];
    #pragma unroll
    for (int i = 1; i < 5; ++i) pm = fmaxf(pm, p2[i]);
    float psum = 0.0f;
    #pragma unroll
    for (int i = 0; i < 5; ++i) { pbuf[i] = expf(p2[i] - pm); psum += pbuf[i]; }
    const float ps0 = pbuf[0] / psum, ps1 = pbuf[1] / psum, ps2 = pbuf[2] / psum;

    for (int i = t; i < SZ; i += 256) {
        const float val = accs[i] * Zinv;
        const float sg  = 1.0f / (1.0f + expf(-val));
        fc[(size_t)b * SZ + i] = sg * val * ps0 + sinf(val) * ps1 + val * ps2;
    }
}

// ---------------------------------------------------------------------------
// Kernel 3: pre = fc @ Wfc2^T + bfc2   (same WMMA fp32 scheme)
// ---------------------------------------------------------------------------
__global__ __launch_bounds__(128) void fc2_gemm(
    const float* __restrict__ fc,
    const float* __restrict__ Wfc2,
    const float* __restrict__ bias,
    float* __restrict__ out)
{
    const int wave = (blockIdx.x * blockDim.x + threadIdx.x) >> 5; // 0..127
    const int lane = threadIdx.x & 31;
    const int ntile = wave;

    const int m  = lane & 15;
    const int kh = lane >> 4;
    const int n  = lane & 15;

    const float* xr0 = fc + m * SZ;
    const float* xr1 = fc + (16 + m) * SZ;
    const float* wr  = Wfc2 + (size_t)(ntile * 16 + n) * SZ;

    v8f c0 = {}; v8f c1 = {};
    #pragma unroll 8
    for (int k0 = 0; k0 < SZ; k0 += 4) {
        const int kk = k0 + kh * 2;
        v2f a0 = *(const v2f*)(xr0 + kk);
        v2f a1 = *(const v2f*)(xr1 + kk);
        v2f bb = *(const v2f*)(wr  + kk);
        c0 = __builtin_amdgcn_wmma_f32_16x16x4_f32(false, a0, false, bb,
                                                   (short)0, c0, false, false);
        c1 = __builtin_amdgcn_wmma_f32_16x16x4_f32(false, a1, false, bb,
                                                   (short)0, c1, false, false);
    }

    const int col = ntile * 16 + n;
    const float bv = bias[col];
    #pragma unroll
    for (int r = 0; r < 8; ++r) {
        out[(size_t)(r + kh * 8) * OUTN + col]      = c0[r] + bv;
        out[(size_t)(16 + r + kh * 8) * OUTN + col] = c1[r] + bv;
    }
}

// ---------------------------------------------------------------------------
// Kernel 4: row LayerNorm -> d_out
// ---------------------------------------------------------------------------
__global__ __launch_bounds__(256) void layernorm_k(
    const float* __restrict__ xin,
    const float* __restrict__ g,
    const float* __restrict__ bb,
    float* __restrict__ out)
{
    __shared__ float red[256];
    __shared__ float stat[2];
    const int b = blockIdx.x;
    const int t = threadIdx.x;
    const float* row = xin + (size_t)b * OUTN;

    float s = 0.0f;
    for (int i = t; i < OUTN; i += 256) s += row[i];
    red[t] = s; __syncthreads();
    if (t == 0) { float tot = 0.0f; for (int i = 0; i < 256; ++i) tot += red[i]; stat[0] = tot / OUTN; }
    __syncthreads();
    const float mu = stat[0];

    float vsum = 0.0f;
    for (int i = t; i < OUTN; i += 256) { const float d = row[i] - mu; vsum += d * d; }
    red[t] = vsum; __syncthreads();
    if (t == 0) { float tot = 0.0f; for (int i = 0; i < 256; ++i) tot += red[i]; stat[1] = rsqrtf(tot / OUTN + LN_EPS); }
    __syncthreads();
    const float rs = stat[1];

    for (int i = t; i < OUTN; i += 256)
        out[(size_t)b * OUTN + i] = (row[i] - mu) * rs * g[i] + bb[i];
}

// ---------------------------------------------------------------------------
extern "C" void kernel_launch(void* const* d_in, const int* in_sizes, int n_in,
                              void* d_out, int out_size, void* d_ws, size_t ws_size,
                              hipStream_t stream)
{
    const float* x    = (const float*)d_in[0];
    const float* Wq   = (const float*)d_in[1];
    const float* Wk   = (const float*)d_in[2];
    const float* Wv   = (const float*)d_in[3];
    const float* p2   = (const float*)d_in[4];
    const float* Wfc2 = (const float*)d_in[5];
    const float* bfc2 = (const float*)d_in[6];
    const float* g2   = (const float*)d_in[7];
    const float* b2   = (const float*)d_in[8];

    float* ws  = (float*)d_ws;
    float* q   = ws;
    float* k   = ws + (size_t)BSZ * SZ;
    float* v   = ws + (size_t)2 * BSZ * SZ;
    float* fc  = ws + (size_t)3 * BSZ * SZ;
    float* pre = ws + (size_t)4 * BSZ * SZ;

    // 3 weights * 128 N-strips = 384 waves -> 96 blocks of 128 threads
    qkv_gemm<<<96, 128, 0, stream>>>(x, Wq, Wk, Wv, q, k, v);
    attn_act<<<BSZ, 256, 0, stream>>>(q, k, v, p2, fc);
    // 128 N-strips = 128 waves -> 32 blocks of 128 threads
    fc2_gemm<<<32, 128, 0, stream>>>(fc, Wfc2, bfc2, pre);
    layernorm_k<<<BSZ, 256, 0, stream>>>(pre, g2, b2, (float*)d_out);
}